// AnchorFreeSingleV2_35407710388836
// MI455X (gfx1250) — compile-verified
//
#include <hip/hip_runtime.h>
#include <hip/hip_bf16.h>
#include <stdint.h>

// ---------------- problem constants (from setup_inputs) ----------------
#define BB   8
#define CC   3
#define HH   496
#define WW   432
#define HWSZ (HH * WW)
#define KK   500

#define ROWS_PER_BLK 16
#define NBLK (HH / ROWS_PER_BLK)      // 31 segments per channel
#define LSZ  512                      // per-block sorted-list length (>= K)

// ---------------- CDNA5 async global->LDS helpers ----------------
__device__ __forceinline__ void async_g2lds_b32(uint32_t lds_byte_off, const float* gaddr) {
  // ASYNCcnt-tracked DMA of one dword per lane into LDS (gfx1250).
  asm volatile("global_load_async_to_lds_b32 %0, %1, off"
               :: "v"(lds_byte_off), "v"(gaddr) : "memory");
}

__device__ __forceinline__ void wait_async0() {
#if __has_builtin(__builtin_amdgcn_s_wait_asynccnt)
  __builtin_amdgcn_s_wait_asynccnt(0);
#else
  asm volatile("s_wait_asynccnt 0x0" ::: "memory");
#endif
}

// ---------------- ordering: score desc, index asc (lax.top_k ties) ----------------
__device__ __forceinline__ bool pgt(float as, int ai, float bs, int bi) {
  return (as > bs) || (as == bs && ai < bi);
}

// Bitonic full sort, descending by (score, idx-asc). n power of two.
__device__ __forceinline__ void bitonic_sort_desc(float* s, int* idx, int tid, int n, int nthr) {
  for (int k = 2; k <= n; k <<= 1) {
    for (int j = k >> 1; j > 0; j >>= 1) {
      for (int i = tid; i < n; i += nthr) {
        int p = i ^ j;
        if (p > i) {
          bool desc_blk = ((i & k) == 0);
          float si = s[i], sp = s[p];
          int   di = idx[i], dp = idx[p];
          bool i_first = pgt(si, di, sp, dp);
          bool do_swap = desc_blk ? !i_first : i_first;
          if (do_swap) {
            s[i]   = sp;  idx[i] = dp;
            s[p]   = si;  idx[p] = di;
          }
        }
      }
      __syncthreads();
    }
  }
}

// Bitonic merge of a bitonic sequence into fully descending order.
__device__ __forceinline__ void bitonic_merge_desc(float* s, int* idx, int tid, int n, int nthr) {
  for (int j = n >> 1; j > 0; j >>= 1) {
    for (int i = tid; i < n; i += nthr) {
      int p = i ^ j;
      if (p > i) {
        float si = s[i], sp = s[p];
        int   di = idx[i], dp = idx[p];
        if (!pgt(si, di, sp, dp)) {
          s[i] = sp; idx[i] = dp;
          s[p] = si; idx[p] = di;
        }
      }
    }
    __syncthreads();
  }
}

// Merge sorted-desc chunk (c) into sorted-desc best (b): top-L of union.
__device__ __forceinline__ void topk_merge(float* bs, int* bi, float* cs, int* ci,
                                           int tid, int n, int nthr) {
  for (int i = tid; i < n; i += nthr) {
    float as_ = bs[i]; int ai_ = bi[i];
    float xs_ = cs[n - 1 - i]; int xi_ = ci[n - 1 - i];
    bool a_first = pgt(as_, ai_, xs_, xi_);
    bs[i] = a_first ? as_ : xs_;               // own-slot write, no cross-thread hazard
    bi[i] = a_first ? ai_ : xi_;
  }
  __syncthreads();
  bitonic_merge_desc(bs, bi, tid, n, nthr);
}

// ================= Kernel 1: sigmoid + clip + 3x3 NMS + segment top-512 =================
__global__ __launch_bounds__(512)
void nms_topk_seg(const float* __restrict__ hm, float* __restrict__ s1, int* __restrict__ i1) {
  __shared__ float rows[(ROWS_PER_BLK + 2) * WW];     // 18*432 floats, raw -> clipped sigmoid
  __shared__ float bs[LSZ]; __shared__ int bi[LSZ];
  __shared__ float cs[LSZ]; __shared__ int ci[LSZ];

  const int tid = threadIdx.x;
  const int blk = blockIdx.x;
  const int seg = blk % NBLK;
  const int bc  = blk / NBLK;                 // fused (b,c)
  const int r0  = seg * ROWS_PER_BLK;
  const float* chan = hm + (size_t)bc * HWSZ;

  // ---- stage 18 rows (halo) into LDS via async DMA; -inf padding rows via DS stores ----
  const int STG = (ROWS_PER_BLK + 2) * WW;    // 7776
  for (int p = tid; p < STG; p += 512) {
    int hh = r0 - 1 + p / WW;
    if (hh >= 0 && hh < HH) {
      uint32_t lo = (uint32_t)(uintptr_t)(&rows[p]);
      async_g2lds_b32(lo, chan + (size_t)hh * WW + (p % WW));
    } else {
      rows[p] = -__builtin_inff();
    }
  }
  wait_async0();
  __syncthreads();

  // ---- in-place clipped sigmoid (keep -inf halo as -inf to match SAME padding) ----
  for (int p = tid; p < STG; p += 512) {
    float x = rows[p];
    if (x > -1e30f) {
      float sg = 1.0f / (1.0f + __expf(-x));
      sg = fminf(fmaxf(sg, 1.0e-4f), 1.0f - 1.0e-4f);
      rows[p] = sg;
    }
  }
  // init best list
  bs[tid] = -1.0f; bi[tid] = 0x7FFFFFFF;
  __syncthreads();

  // ---- scan 16*432 pixels in 512-wide chunks: NMS score -> bitonic top-512 ----
  const int E = ROWS_PER_BLK * WW;            // 6912
  for (int c0 = 0; c0 < E; c0 += 512) {
    int p = c0 + tid;
    float sc = -1.0f; int gi = 0x7FFFFFFF;
    if (p < E) {
      int lr = p / WW + 1;                    // halo offset
      int lc = p % WW;
      int base = lr * WW;
      float v = rows[base + lc];
      float m = v;
      m = fmaxf(m, rows[base - WW + lc]);
      m = fmaxf(m, rows[base + WW + lc]);
      if (lc > 0) {
        m = fmaxf(m, rows[base - WW + lc - 1]);
        m = fmaxf(m, rows[base      + lc - 1]);
        m = fmaxf(m, rows[base + WW + lc - 1]);
      }
      if (lc < WW - 1) {
        m = fmaxf(m, rows[base - WW + lc + 1]);
        m = fmaxf(m, rows[base      + lc + 1]);
        m = fmaxf(m, rows[base + WW + lc + 1]);
      }
      sc = (m == v) ? v : 0.0f;               // heat * keep
      gi = (r0 + p / WW) * WW + lc;           // flat H*W index
    }
    cs[tid] = sc; ci[tid] = gi;
    __syncthreads();
    bitonic_sort_desc(cs, ci, tid, LSZ, 512);
    topk_merge(bs, bi, cs, ci, tid, LSZ, 512);
  }

  s1[(size_t)blk * LSZ + tid] = bs[tid];
  i1[(size_t)blk * LSZ + tid] = bi[tid];
}

// ================= Kernel 2: per-(b,c) merge of 31 sorted lists -> top-512 =================
__global__ __launch_bounds__(512)
void merge_channel(const float* __restrict__ s1, const int* __restrict__ i1,
                   float* __restrict__ s2, int* __restrict__ i2) {
  __shared__ float bs[LSZ]; __shared__ int bi[LSZ];
  __shared__ float cs[LSZ]; __shared__ int ci[LSZ];
  const int tid = threadIdx.x;
  const int bc  = blockIdx.x;

  bs[tid] = -1.0f; bi[tid] = 0x7FFFFFFF;
  __syncthreads();
  for (int blk = 0; blk < NBLK; ++blk) {
    size_t o = ((size_t)bc * NBLK + blk) * LSZ + tid;
    cs[tid] = s1[o]; ci[tid] = i1[o];
    __syncthreads();
    topk_merge(bs, bi, cs, ci, tid, LSZ, 512);  // chunks already sorted desc
  }
  s2[(size_t)bc * LSZ + tid] = bs[tid];
  i2[(size_t)bc * LSZ + tid] = bi[tid];
}

// ================= Kernel 3: per-batch top-500 over C*K + feature gather =================
__global__ __launch_bounds__(1024)
void final_topk_gather(const float* __restrict__ s2, const int* __restrict__ i2,
                       const float* __restrict__ cen_offset,
                       const float* __restrict__ direction,
                       const float* __restrict__ z_coor,
                       const float* __restrict__ dimf,
                       float* __restrict__ out) {
  __shared__ float ss[2048]; __shared__ int jj[2048];
  const int tid = threadIdx.x;
  const int b   = blockIdx.x;

  for (int i = tid; i < 2048; i += 1024) {
    if (i < CC * KK) {
      int cls = i / KK, rank = i % KK;
      ss[i] = s2[((size_t)b * CC + cls) * LSZ + rank];
      jj[i] = i;                               // == cls*K + rank: reference's (C*K) index
    } else {
      ss[i] = -1.0f; jj[i] = 0x7FFFFFFF;
    }
  }
  __syncthreads();
  bitonic_sort_desc(ss, jj, tid, 2048, 1024);

  if (tid < KK) {
    float score = ss[tid];
    int j = jj[tid];
    int cls = j / KK, rank = j % KK;
    int ind = i2[((size_t)b * CC + cls) * LSZ + rank];
    if (ind == 0x7FFFFFFF) ind = 0;            // defensive (never hit: >=24k peaks/channel)
    float row = (float)(ind / WW);
    float col = (float)(ind % WW);

    const float* offp = cen_offset + (size_t)b * 2 * HWSZ;
    float ox = offp[ind];
    float oy = offp[HWSZ + ind];
    // reference applies _sigmoid (clipped) to cen_offset before the gather
    ox = fminf(fmaxf(1.0f / (1.0f + __expf(-ox)), 1.0e-4f), 1.0f - 1.0e-4f);
    oy = fminf(fmaxf(1.0f / (1.0f + __expf(-oy)), 1.0e-4f), 1.0f - 1.0e-4f);

    const float* dirp = direction + (size_t)b * 2 * HWSZ;
    float d0 = dirp[ind], d1 = dirp[HWSZ + ind];
    float z  = z_coor[(size_t)b * HWSZ + ind];
    const float* dmp = dimf + (size_t)b * 3 * HWSZ;
    float m0 = dmp[ind], m1 = dmp[HWSZ + ind], m2 = dmp[2 * HWSZ + ind];

    float* o = out + ((size_t)b * KK + tid) * 10;
    o[0] = score;
    o[1] = col + ox;
    o[2] = row + oy;
    o[3] = z;
    o[4] = m0; o[5] = m1; o[6] = m2;
    o[7] = d0; o[8] = d1;
    o[9] = (float)cls;
  }
}

// ================= host launcher =================
extern "C" void kernel_launch(void* const* d_in, const int* in_sizes, int n_in,
                              void* d_out, int out_size, void* d_ws, size_t ws_size,
                              hipStream_t stream) {
  (void)in_sizes; (void)n_in; (void)out_size; (void)ws_size;
  const float* hm   = (const float*)d_in[0];
  const float* off  = (const float*)d_in[1];
  const float* dir  = (const float*)d_in[2];
  const float* z    = (const float*)d_in[3];
  const float* dimf = (const float*)d_in[4];
  // d_in[5] is K (==500), baked in as KK.

  const size_t S1 = (size_t)BB * CC * NBLK * LSZ;   // 380928 elems
  const size_t S2 = (size_t)BB * CC * LSZ;          // 12288 elems
  float* s1 = (float*)d_ws;
  int*   i1 = (int*)(s1 + S1);
  float* s2 = (float*)(i1 + S1);
  int*   i2 = (int*)(s2 + S2);

  nms_topk_seg<<<BB * CC * NBLK, 512, 0, stream>>>(hm, s1, i1);
  merge_channel<<<BB * CC, 512, 0, stream>>>(s1, i1, s2, i2);
  final_topk_gather<<<BB, 1024, 0, stream>>>(s2, i2, off, dir, z, dimf, (float*)d_out);
}